// Gate_38792144617563
// MI455X (gfx1250) — compile-verified
//
#include <hip/hip_runtime.h>
#include <math.h>

#define B_ 4
#define S_ 4096
#define H_ 2048
#define E_ 8
#define K_ 8
#define T_ (B_ * S_)          // 16384 tokens
#define SEQ_ALPHA 0.01f

typedef __attribute__((ext_vector_type(2))) float v2f;
typedef __attribute__((ext_vector_type(8))) float v8f;

// One wave (wave32) computes logits for a 16-token tile against all 8 experts
// using V_WMMA_F32_16X16X4_F32. Block = 256 threads = 8 waves = 128 tokens.
__global__ __launch_bounds__(256) void gate_wmma_kernel(
    const float* __restrict__ x,            // [T, H]
    const float* __restrict__ weight,       // [E, H]
    const float* __restrict__ expert_bias,  // [E]
    float* __restrict__ out,                // concatenated outputs
    float* __restrict__ gP,                 // [B, E] normalized-score sums
    float* __restrict__ gC)                 // [B, E] selection counts
{
    __shared__ float s_logits[8][16 * E_];  // per-wave 16 tokens x 8 experts
    __shared__ float sP[E_];
    __shared__ float sC[E_];

    const int tid  = threadIdx.x;
    const int wave = tid >> 5;
    const int lane = tid & 31;
    const int hh   = lane >> 4;   // half of wave: selects K-pair (and M/N half of D)
    const int lm   = lane & 15;   // A: M row; B: N column

    if (tid < E_) { sP[tid] = 0.0f; sC[tid] = 0.0f; }

    const long tile = (long)blockIdx.x * 8 + wave;            // 16-token tile id
    const float* xrow = x + (tile * 16 + lm) * (long)H_ + hh * 2;
    const float* wrow = weight + (long)(lm & 7) * H_ + hh * 2; // N>=8 duplicates N-8; ignored

    v8f acc0 = {0.f, 0.f, 0.f, 0.f, 0.f, 0.f, 0.f, 0.f};
    v8f acc1 = {0.f, 0.f, 0.f, 0.f, 0.f, 0.f, 0.f, 0.f};

    for (int k = 0; k < H_; k += 8) {
        v2f a0 = *(const v2f*)(xrow + k);
        v2f b0 = *(const v2f*)(wrow + k);
        v2f a1 = *(const v2f*)(xrow + k + 4);
        v2f b1 = *(const v2f*)(wrow + k + 4);
        // 8 args: (neg_a, A, neg_b, B, c_mod, C, reuse_a, reuse_b)
        acc0 = __builtin_amdgcn_wmma_f32_16x16x4_f32(
            false, a0, false, b0, (short)0, acc0, false, false);
        acc1 = __builtin_amdgcn_wmma_f32_16x16x4_f32(
            false, a1, false, b1, (short)0, acc1, false, false);
    }
    v8f acc = acc0 + acc1;

    // D layout: lanes 0-15 hold D[v][lane], lanes 16-31 hold D[v+8][lane-16].
    // Only experts (columns) 0..7 are meaningful.
    if (lm < 8) {
#pragma unroll
        for (int v = 0; v < 8; ++v)
            s_logits[wave][(v + hh * 8) * E_ + lm] = acc[v];
    }
    __syncthreads();

    if (tid < 128) {  // one thread per token of this block
        const int  w     = tid >> 4;
        const int  m     = tid & 15;
        const long token = (long)blockIdx.x * 128 + tid;

        float lg[E_], sc[E_], bl[E_];
        float ssum = 0.0f;
#pragma unroll
        for (int e = 0; e < E_; ++e) {
            lg[e] = s_logits[w][m * E_ + e];
            sc[e] = 1.0f / (1.0f + __expf(-lg[e]));
            ssum += sc[e];
            bl[e] = lg[e] + expert_bias[e];
        }

        // Stable descending selection (matches lax.top_k tie-breaking).
        int      order[K_];
        unsigned used  = 0u;
        float    twsum = 0.0f;
        for (int j = 0; j < K_; ++j) {
            int best = 0; float bv = -INFINITY;
#pragma unroll
            for (int e = 0; e < E_; ++e)
                if (!(used & (1u << e)) && bl[e] > bv) { bv = bl[e]; best = e; }
            used |= 1u << best;
            order[j] = best;
            twsum += sc[best];
        }

        const float inv_t = 1.0f / (twsum + 1e-10f);
        const float inv_s = 1.0f / (ssum + 1e-10f);
#pragma unroll
        for (int j = 0; j < K_; ++j) {
            out[token * K_ + j]                   = (float)order[j];
            out[(long)T_ * K_ + token * K_ + j]   = sc[order[j]] * inv_t;
            atomicAdd(&sC[order[j]], 1.0f);
        }
#pragma unroll
        for (int e = 0; e < E_; ++e)
            atomicAdd(&sP[e], sc[e] * inv_s);
    }
    __syncthreads();

    if (tid < E_) {
        const int b = (int)(((long)blockIdx.x * 128) / S_);  // whole block in one batch
        atomicAdd(&gP[b * E_ + tid], sP[tid]);
        atomicAdd(&gC[b * E_ + tid], sC[tid]);
    }
}

__global__ void gate_finalize_kernel(const float* __restrict__ gP,
                                     const float* __restrict__ gC,
                                     float* __restrict__ out)
{
    const int t = threadIdx.x;
    if (t == 0) {
        float s = 0.0f;
        for (int b = 0; b < B_; ++b)
            for (int e = 0; e < E_; ++e)
                s += (gC[b * E_ + e] / (float)(K_ * S_)) * (gP[b * E_ + e] / (float)S_);
        out[2L * T_ * K_] = SEQ_ALPHA * s / (float)B_;
    }
    if (t < E_) {
        float l = 0.0f;
        for (int b = 0; b < B_; ++b) l += gC[b * E_ + t];
        out[2L * T_ * K_ + 1 + t] = l;
    }
}

extern "C" void kernel_launch(void* const* d_in, const int* in_sizes, int n_in,
                              void* d_out, int out_size, void* d_ws, size_t ws_size,
                              hipStream_t stream) {
    const float* x      = (const float*)d_in[0];
    const float* weight = (const float*)d_in[1];
    const float* bias   = (const float*)d_in[2];
    float* out = (float*)d_out;
    float* gP  = (float*)d_ws;
    float* gC  = gP + B_ * E_;

    hipMemsetAsync(d_ws, 0, 2 * B_ * E_ * sizeof(float), stream);
    gate_wmma_kernel<<<T_ / 128, 256, 0, stream>>>(x, weight, bias, out, gP, gC);
    gate_finalize_kernel<<<1, 32, 0, stream>>>(gP, gC, out);
}